// YOLOLoss_35596688949531
// MI455X (gfx1250) — compile-verified
//
#include <hip/hip_runtime.h>
#include <math.h>

// ---------------- constants (YOLOv3 config; head l=2, 512/64 => scale 8) ----
#define BS   16
#define NT   20
#define HH   64
#define WW   64
#define NA   3
#define NC   80
#define MTOT (BS * NA * HH * WW)       // 196608 cells
#define NBLK (MTOT / 256)              // 768 blocks, exactly 1 thread/cell

__constant__ float c_anchors[9][2] = {
    {10.f,13.f},{16.f,30.f},{33.f,23.f},{30.f,61.f},{62.f,45.f},
    {59.f,119.f},{116.f,90.f},{156.f,198.f},{373.f,326.f}};
__constant__ int c_place[3][3] = {{6,7,8},{3,4,5},{0,1,2}};

typedef __attribute__((ext_vector_type(2))) float v2f;
typedef __attribute__((ext_vector_type(8))) float v8f;

// min(softplus(z),100): equals -max(log(sigmoid(±z)),-100) terms of reference
__device__ __forceinline__ float spc(float z) {
    float v = fmaxf(z, 0.f) + log1pf(expf(-fabsf(z)));
    return fminf(v, 100.f);
}
// BCE(sigmoid(z), t) with torch-style -100 log clamp
__device__ __forceinline__ float bce(float z, float t) {
    return t * spc(-z) + (1.f - t) * spc(z);
}
__device__ __forceinline__ float sigm(float z) { return 1.f / (1.f + expf(-z)); }

// ---------------------------------------------------------------- kernel 1 --
__global__ void k_targets(const float* __restrict__ txywh,
                          const int* __restrict__ tcls,
                          const int* __restrict__ lptr,
                          float* __restrict__ tgtVals,   // [320*6]
                          float* __restrict__ tgtBox,    // [320*4]
                          int*   __restrict__ tgtCell)   // [320]
{
    int i = threadIdx.x;
    if (i >= BS * NT) return;
    int l = *lptr; l = l < 0 ? 0 : (l > 2 ? 2 : l);
    int b = i / NT, t = i % NT;
    const float* p = txywh + (size_t)(b * NT + t) * 4;
    float tx = p[0] * WW, ty = p[1] * HH, tw = p[2] * WW, th = p[3] * HH;
    float areaT = tw * th;

    // anchor match on (w,h) only; anchors in grid units (pixel/8); first-max
    int best = 0; float bestIou = -1.f;
    for (int a = 0; a < 9; ++a) {
        float aw = c_anchors[a][0] * 0.125f, ah = c_anchors[a][1] * 0.125f;
        float inter = fminf(tw, aw) * fminf(th, ah);
        float iou = inter / (areaT + aw * ah - inter);
        if (iou > bestIou) { bestIou = iou; best = a; }
    }
    int k = 0; bool valid = false;
    for (int kk = 2; kk >= 0; --kk)                 // first match wins
        if (c_place[l][kk] == best) { k = kk; valid = true; }

    int gi = (int)tx, gj = (int)ty;                 // trunc == floor (tx,ty>0)
    int cell = ((b * NA + k) * HH + gj) * WW + gi;
    tgtCell[i] = valid ? cell : -1;

    float aw = c_anchors[best][0] * 0.125f, ah = c_anchors[best][1] * 0.125f;
    tgtVals[i*6+0] = tx - (float)gi;
    tgtVals[i*6+1] = ty - (float)gj;
    tgtVals[i*6+2] = logf(tw / aw);
    tgtVals[i*6+3] = logf(th / ah);
    tgtVals[i*6+4] = areaT * (1.f / (float)(WW * HH));
    tgtVals[i*6+5] = (float)tcls[b * NT + t];
    tgtBox[i*4+0] = tx - tw * 0.5f;
    tgtBox[i*4+1] = ty - th * 0.5f;
    tgtBox[i*4+2] = tx + tw * 0.5f;
    tgtBox[i*4+3] = ty + th * 0.5f;
}

// ---------------------------------------------------------------- kernel 2 --
// One thread per cell; 9 partial sums; WMMA cross-wave block reduction.
__global__ void __launch_bounds__(256)
k_main(const float* __restrict__ x, const int* __restrict__ lptr,
       const float* __restrict__ tgtVals, const float* __restrict__ tgtBox,
       const int* __restrict__ tgtCell, float* __restrict__ blockSums)
{
    int m = blockIdx.x * 256 + threadIdx.x;       // grid == MTOT exactly
    int j = m & 63;
    int i = (m >> 6) & 63;
    int a = (m >> 12) % NA;
    int b = m / (NA * HH * WW);
    int l = *lptr; l = l < 0 ? 0 : (l > 2 ? 2 : l);

    const float* base = x + (size_t)(b * 255 + a * 85) * 4096 + i * 64 + j;
    float z0 = base[0];
    float z1 = base[4096];
    float z2 = base[2 * 4096];
    float z3 = base[3 * 4096];
    float z4 = base[4 * 4096];

    float s0 = sigm(z0), s1 = sigm(z1);
    float aw = c_anchors[c_place[l][a]][0] * 0.125f;
    float ah = c_anchors[c_place[l][a]][1] * 0.125f;
    float pw = expf(z2) * aw, ph = expf(z3) * ah;
    float px = s0 + (float)j, py = s1 + (float)i;
    float px1 = px - pw * 0.5f, py1 = py - ph * 0.5f;
    float px2 = px + pw * 0.5f, py2 = py + ph * 0.5f;
    float areaP = (px2 - px1) * (py2 - py1);

    int assigned = -1; bool sup = false;
    #pragma unroll 4
    for (int t = 0; t < NT; ++t) {
        int ti = b * NT + t;
        int ci = tgtCell[ti];
        if (ci == m) assigned = ti;               // last writer wins (scatter)
        if (ci >= 0) {                            // valid targets suppress
            float tx1 = tgtBox[ti*4+0], ty1 = tgtBox[ti*4+1];
            float tx2 = tgtBox[ti*4+2], ty2 = tgtBox[ti*4+3];
            float iw = fmaxf(fminf(tx2, px2) - fmaxf(tx1, px1), 0.f);
            float ih = fmaxf(fminf(ty2, py2) - fmaxf(ty1, py1), 0.f);
            float inter = iw * ih;
            float areaT = (tx2 - tx1) * (ty2 - ty1);
            float iou = inter / (areaT + areaP - inter);
            if (iou > 0.5f) sup = true;
        }
    }

    float tX = 0.f, tY = 0.f, tW = 0.f, tH = 0.f, sw = 0.f, obj = 0.f;
    int cls = -1;
    if (assigned >= 0) {
        tX = tgtVals[assigned*6+0]; tY = tgtVals[assigned*6+1];
        tW = tgtVals[assigned*6+2]; tH = tgtVals[assigned*6+3];
        sw = tgtVals[assigned*6+4]; cls = (int)tgtVals[assigned*6+5];
        obj = 1.f;
    }

    float s[9];
    s[0] = bce(z0, tX);
    s[1] = bce(z1, tY);
    s[2] = (z2 - tW) * (z2 - tW);
    s[3] = (z3 - tH) * (z3 - tH);
    s[4] = bce(z4, obj);
    s[5] = obj;
    s[6] = (assigned >= 0 || sup) ? 0.f : 1.f;    // noObj
    s[7] = (assigned >= 0) ? (2.f - sw) : 0.f;    // obj * smallWeight
    s[8] = 0.f;                                   // class BCE (assigned only)
    if (assigned >= 0) {
        for (int c = 0; c < NC; ++c) {
            float zc = base[(5 + c) * 4096];
            s[8] += bce(zc, (c == cls) ? 1.f : 0.f);
        }
    }

    // intra-wave reduction (wave32)
    #pragma unroll
    for (int off = 16; off; off >>= 1)
        #pragma unroll
        for (int q = 0; q < 9; ++q)
            s[q] += __shfl_xor(s[q], off, 32);

    __shared__ float S[8][16];                    // waves x sums (padded)
    int wave = threadIdx.x >> 5, lane = threadIdx.x & 31;
    if (lane == 0) {
        #pragma unroll
        for (int q = 0; q < 9; ++q) S[wave][q] = s[q];
        #pragma unroll
        for (int q = 9; q < 16; ++q) S[wave][q] = 0.f;
    }
    __syncthreads();

    // Cross-wave reduction via V_WMMA_F32_16X16X4_F32: D = ones(16x4) x B(4x16) + C.
    // Two chained WMMAs cover the 8 wave-rows; every D element in column n is
    // the exact column sum, invariant to K/M row permutations (A is all-ones).
    // EXEC is all-ones: full block, no divergence here.
    int col  = lane & 15;
    int half = lane >> 4;
    v2f A;  A.x = 1.f; A.y = 1.f;
    v2f B0; B0.x = S[2 * half + 0][col]; B0.y = S[2 * half + 1][col];
    v2f B1; B1.x = S[2 * half + 4][col]; B1.y = S[2 * half + 5][col];
    v8f D = {};
    D = __builtin_amdgcn_wmma_f32_16x16x4_f32(false, A, false, B0, (short)0, D, false, false);
    D = __builtin_amdgcn_wmma_f32_16x16x4_f32(false, A, false, B1, (short)0, D, false, false);

    if (threadIdx.x < 16)
        blockSums[blockIdx.x * 16 + threadIdx.x] = D[0];
}

// ---------------------------------------------------------------- kernel 3 --
__global__ void k_final(const float* __restrict__ blockSums,
                        float* __restrict__ out)
{
    __shared__ float tot[9];
    int wave = threadIdx.x >> 5, lane = threadIdx.x & 31;
    if (wave < 9) {
        float acc = 0.f;
        for (int blk = lane; blk < NBLK; blk += 32)
            acc += blockSums[blk * 16 + wave];
        #pragma unroll
        for (int off = 16; off; off >>= 1)
            acc += __shfl_xor(acc, off, 32);
        if (lane == 0) tot[wave] = acc;
    }
    __syncthreads();
    if (threadIdx.x == 0) {
        const float Mf = (float)MTOT;
        float sX = tot[0], sY = tot[1], sW = tot[2], sH = tot[3], sP = tot[4];
        float obj = tot[5], noObj = tot[6], objSW = tot[7], sC = tot[8];
        float lossX = (sX / Mf) * objSW;
        float lossY = (sY / Mf) * objSW;
        float lossW = (sW / Mf) * (objSW * 0.5f);
        float lossH = (sH / Mf) * (objSW * 0.5f);
        float bceP  = sP / Mf;
        float lossP = bceP * obj + bceP * noObj;
        float lossC = (obj > 0.f) ? sC / (obj * (float)NC) : 0.f;
        out[0] = lossX + lossY + lossW + lossH + lossP + lossC;
        out[1] = fmaxf(obj, 1.f);
    }
}

// ------------------------------------------------------------------ launch --
extern "C" void kernel_launch(void* const* d_in, const int* in_sizes, int n_in,
                              void* d_out, int out_size, void* d_ws, size_t ws_size,
                              hipStream_t stream)
{
    const float* x     = (const float*)d_in[0];
    const float* txywh = (const float*)d_in[1];
    const int*   tcls  = (const int*)d_in[2];
    const int*   lptr  = (const int*)d_in[3];
    float* out = (float*)d_out;

    // workspace layout (all regions fully written before read)
    float* tgtVals   = (float*)d_ws;                 // 320*6 floats
    float* tgtBox    = tgtVals + BS * NT * 6;        // 320*4 floats
    int*   tgtCell   = (int*)(tgtBox + BS * NT * 4); // 320 ints
    float* blockSums = (float*)(tgtCell + BS * NT);  // 768*16 floats

    k_targets<<<1, BS * NT, 0, stream>>>(txywh, tcls, lptr, tgtVals, tgtBox, tgtCell);
    k_main<<<NBLK, 256, 0, stream>>>(x, lptr, tgtVals, tgtBox, tgtCell, blockSums);
    k_final<<<1, 288, 0, stream>>>(blockSums, out);
}